// hinge_onevsrest_v0_64244120813576
// MI455X (gfx1250) — compile-verified
//
#include <hip/hip_runtime.h>
#include <hip/hip_bf16.h>

typedef __attribute__((ext_vector_type(2))) float v2f;
typedef __attribute__((ext_vector_type(4))) float v4f;
typedef __attribute__((ext_vector_type(8))) float v8f;

#define NBLOCKS 2048
#define NTHREADS 256
#define BATCH_F 16384.0f

// Per-element hinge: sign = (label >= 0) ? +1 : -1 ; h = max(1 - out*sign, 0)
__device__ __forceinline__ float hinge_elem(float o, float l) {
    float so = (l >= 0.0f) ? o : -o;
    return fmaxf(1.0f - so, 0.0f);
}

__device__ __forceinline__ float hinge4(v4f o, v4f l) {
    float a = hinge_elem(o[0], l[0]) + hinge_elem(o[1], l[1]);
    float b = hinge_elem(o[2], l[2]) + hinge_elem(o[3], l[3]);
    return a + b;
}

// Wave-level sum of per-lane partial `p` using V_WMMA_F32_16X16X4_F32.
// A (16x4 f32, 2 VGPRs/lane): lane L holds A[L%16][2*(L/16) + {0,1}].
// p in slot 0, 0 in slot 1; B = all ones (4x16) => D[m][n] = rowsum_m.
// Lane n sums its 8 C regs (rows 0..7 or 8..15 of column n); adding the
// partner lane via shfl_xor(16) yields the full wave sum in every lane.
// Requires EXEC all-ones (call with whole wave active, no divergence).
__device__ __forceinline__ float wave_reduce_wmma(float p) {
    v2f a; a[0] = p;    a[1] = 0.0f;
    v2f b; b[0] = 1.0f; b[1] = 1.0f;
    v8f c = {};
    c = __builtin_amdgcn_wmma_f32_16x16x4_f32(
        /*neg_a=*/false, a, /*neg_b=*/false, b,
        /*c_mod=*/(short)0, c, /*reuse_a=*/false, /*reuse_b=*/false);
    float s = ((c[0] + c[1]) + (c[2] + c[3])) + ((c[4] + c[5]) + (c[6] + c[7]));
    s += __shfl_xor(s, 16, 32);
    return s;
}

// Block-level reduction: WMMA wave reduce, then fixed-order LDS combine.
__device__ __forceinline__ void block_reduce_store(float p, float* dst, float scale) {
    __shared__ float wsum[NTHREADS / 32];
    const int lane = threadIdx.x & 31;
    const int wid  = threadIdx.x >> 5;
    float w = wave_reduce_wmma(p);
    if (lane == 0) wsum[wid] = w;
    __syncthreads();
    if (threadIdx.x == 0) {
        float t = 0.0f;
        #pragma unroll
        for (int i = 0; i < NTHREADS / 32; ++i) t += wsum[i];
        *dst = t * scale;
    }
}

__global__ void hinge_partial_kernel(const float* __restrict__ outputs,
                                     const float* __restrict__ labels,
                                     float* __restrict__ partials,
                                     int n4, int n) {
    const int tid     = blockIdx.x * blockDim.x + threadIdx.x;
    const int nthread = gridDim.x * blockDim.x;
    const v4f* __restrict__ o4 = (const v4f*)outputs;
    const v4f* __restrict__ l4 = (const v4f*)labels;

    // 2x unrolled grid-stride loop: 4 independent NT b128 loads in flight
    // per iteration, two independent accumulators.
    float p0 = 0.0f, p1 = 0.0f;
    int i = tid;
    for (; i + nthread < n4; i += 2 * nthread) {
        v4f oa = __builtin_nontemporal_load(&o4[i]);
        v4f la = __builtin_nontemporal_load(&l4[i]);
        v4f ob = __builtin_nontemporal_load(&o4[i + nthread]);
        v4f lb = __builtin_nontemporal_load(&l4[i + nthread]);
        p0 += hinge4(oa, la);
        p1 += hinge4(ob, lb);
    }
    if (i < n4) {
        v4f oa = __builtin_nontemporal_load(&o4[i]);
        v4f la = __builtin_nontemporal_load(&l4[i]);
        p0 += hinge4(oa, la);
    }
    float p = p0 + p1;

    // Scalar tail (empty for 16,384,000 elements, kept for generality).
    if (blockIdx.x == 0) {
        for (int k = n4 * 4 + threadIdx.x; k < n; k += blockDim.x)
            p += hinge_elem(outputs[k], labels[k]);
    }
    block_reduce_store(p, &partials[blockIdx.x], 1.0f);
}

__global__ void hinge_final_kernel(const float* __restrict__ partials,
                                   float* __restrict__ out,
                                   int nparts) {
    float p = 0.0f;
    for (int i = threadIdx.x; i < nparts; i += blockDim.x)
        p += partials[i];
    block_reduce_store(p, out, 1.0f / BATCH_F);
}

extern "C" void kernel_launch(void* const* d_in, const int* in_sizes, int n_in,
                              void* d_out, int out_size, void* d_ws, size_t ws_size,
                              hipStream_t stream) {
    const float* outputs = (const float*)d_in[0];
    const float* labels  = (const float*)d_in[1];
    float* out           = (float*)d_out;
    float* partials      = (float*)d_ws;   // NBLOCKS floats (8 KB)

    const int n  = in_sizes[0];
    const int n4 = n / 4;

    hinge_partial_kernel<<<NBLOCKS, NTHREADS, 0, stream>>>(outputs, labels, partials, n4, n);
    hinge_final_kernel<<<1, NTHREADS, 0, stream>>>(partials, out, NBLOCKS);
}